// BceLogitsLossWeighted_16063177687049
// MI455X (gfx1250) — compile-verified
//
#include <hip/hip_runtime.h>
#include <math.h>

// ---------------------------------------------------------------------------
// Problem constants (from reference): input/target (N, L, C) f32, C-major rows.
// ---------------------------------------------------------------------------
#define N_BATCH   8
#define LROWS     8192
#define NCH       3
#define SPLITS    16                       // blocks per batch element
#define TPB       256                      // 8 waves (wave32)
#define NBLOCKS   (N_BATCH * SPLITS)       // 128
#define ROWS_PER_BLOCK (LROWS / SPLITS)    // 512
#define NWAVES    (TPB / 32)
#define NQ        11                       // accumulated quantities per block

#define SPARSITY_A 0.1f
#define SPARSITY_B 0.5f

typedef float v2f __attribute__((ext_vector_type(2)));
typedef float v8f __attribute__((ext_vector_type(8)));

// ---------------------------------------------------------------------------
// Exact f32 wave32 sum using V_WMMA_F32_16X16X4_F32 with B = ones.
// A layout (16x4 f32): VGPR0 = {lanes 0-15: K=0, lanes 16-31: K=2},
//                      VGPR1 = {lanes 0-15: K=1, lanes 16-31: K=3}.
// With A.vgpr0 = v, A.vgpr1 = 0, B = 1:  D[m][n] = v[m] + v[16+m] (all n).
// D layout: lane holds rows (lane<16 ? 0..7 : 8..15) across 8 VGPRs, so
// sum of D VGPRs + xor-16 shuffle gives the full 32-lane sum, exactly.
// Requires EXEC all-ones: called only in fully convergent code below.
// ---------------------------------------------------------------------------
__device__ __forceinline__ float wave_sum32(float v) {
    v2f a; a.x = v;   a.y = 0.0f;
    v2f b; b.x = 1.0f; b.y = 1.0f;
    v8f c = {0.0f, 0.0f, 0.0f, 0.0f, 0.0f, 0.0f, 0.0f, 0.0f};
    v8f d = __builtin_amdgcn_wmma_f32_16x16x4_f32(
        /*neg_a=*/false, a, /*neg_b=*/false, b,
        /*c_mod=*/(short)0, c, /*reuse_a=*/false, /*reuse_b=*/false);
    float s = ((d[0] + d[1]) + (d[2] + d[3])) + ((d[4] + d[5]) + (d[6] + d[7]));
    s += __shfl_xor(s, 16, 32);
    return s;
}

// Numerically stable weighted BCE-with-logits term for one element.
__device__ __forceinline__ float bce_term(float x, float t, float pwc) {
    float ax  = fabsf(x);
    float l1  = log1pf(expf(-ax));          // softplus(-|x|)
    float lsp = fminf(x, 0.0f) - l1;        // log_sigmoid(x)
    float lsn = fminf(-x, 0.0f) - l1;       // log_sigmoid(-x)
    return -(pwc * t * lsp + (1.0f - t) * lsn);
}

__device__ __forceinline__ float sp_term(float d) {
    return fabsf(d) - SPARSITY_B * fmaxf(d, 0.0f);
}

// ---------------------------------------------------------------------------
// Pass 1: per-block partial sums.
// Quantities: 0..2 bce[c] (global), 3/4 sum(ch1/ch2), 5/6 sumsq(ch1/ch2),
//             7/8 sparsity(input ch1/ch2), 9/10 sparsity(target ch1/ch2).
// ws layout: ws[q * NBLOCKS + block].
// ---------------------------------------------------------------------------
__global__ __launch_bounds__(TPB)
void loss_partials_kernel(const float* __restrict__ inp,
                          const float* __restrict__ tgt,
                          const float* __restrict__ pw,
                          float* __restrict__ ws) {
    const int n   = blockIdx.x / SPLITS;
    const int seg = blockIdx.x % SPLITS;

    const float pw0 = pw[0], pw1 = pw[1], pw2 = pw[2];

    const size_t batch_base = (size_t)n * LROWS * NCH;
    // First-row broadcast values for the sparsity diffs (channels 1,2).
    const float x0c1 = inp[batch_base + 1];
    const float x0c2 = inp[batch_base + 2];
    const float t0c1 = tgt[batch_base + 1];
    const float t0c2 = tgt[batch_base + 2];

    float acc[NQ];
#pragma unroll
    for (int q = 0; q < NQ; ++q) acc[q] = 0.0f;

    const int row_end = (seg + 1) * ROWS_PER_BLOCK;
    for (int l = seg * ROWS_PER_BLOCK + threadIdx.x; l < row_end; l += TPB) {
        const float* pi = inp + batch_base + (size_t)l * NCH;
        const float* pt = tgt + batch_base + (size_t)l * NCH;
        float x0v = pi[0], x1v = pi[1], x2v = pi[2];
        float t0v = pt[0], t1v = pt[1], t2v = pt[2];

        // BCE per channel
        acc[0] += bce_term(x0v, t0v, pw0);
        acc[1] += bce_term(x1v, t1v, pw1);
        acc[2] += bce_term(x2v, t2v, pw2);

        // Proximity moments (closed form for mean of full xcorr)
        acc[3] += x1v;
        acc[4] += x2v;
        acc[5] += x1v * x1v;
        acc[6] += x2v * x2v;

        // Sparsity (l==0 contributes exactly 0, so no branch)
        acc[7]  += sp_term(x1v - x0c1);
        acc[8]  += sp_term(x2v - x0c2);
        acc[9]  += sp_term(t1v - t0c1);
        acc[10] += sp_term(t2v - t0c2);
    }

    // Wave-level exact reduction via WMMA (all lanes active, EXEC == ~0).
    __shared__ float red[NWAVES][NQ];
    const int wave = threadIdx.x >> 5;
    const int lane = threadIdx.x & 31;
#pragma unroll
    for (int q = 0; q < NQ; ++q) {
        float s = wave_sum32(acc[q]);
        if (lane == 0) red[wave][q] = s;
    }
    __syncthreads();

    // Fixed-order cross-wave combine; deterministic.
    if (threadIdx.x < NQ) {
        float t = 0.0f;
#pragma unroll
        for (int w = 0; w < NWAVES; ++w) t += red[w][threadIdx.x];
        ws[threadIdx.x * NBLOCKS + blockIdx.x] = t;
    }
}

// ---------------------------------------------------------------------------
// Pass 2: tiny finalize, single thread, fixed summation order (deterministic).
// ---------------------------------------------------------------------------
__global__ void loss_finalize_kernel(const float* __restrict__ ws,
                                     const float* __restrict__ cw,
                                     float* __restrict__ out) {
    if (threadIdx.x != 0 || blockIdx.x != 0) return;

    float loss = 0.0f;

    // BCE: mean over N*L per channel, weighted by 10 * class_weights.
    const float inv_nl = 1.0f / (float)(N_BATCH * LROWS);
    for (int c = 0; c < NCH; ++c) {
        float s = 0.0f;
        for (int b = 0; b < NBLOCKS; ++b) s += ws[c * NBLOCKS + b];
        loss += 10.0f * cw[c] * (s * inv_nl);
    }

    // Per-batch proximity + sparsity terms.
    const float inv_len = 1.0f / (2.0f * (float)LROWS - 1.0f);
    for (int n = 0; n < N_BATCH; ++n) {
        float q[NQ - 3];
        for (int k = 0; k < NQ - 3; ++k) {
            float s = 0.0f;
            const float* p = ws + (k + 3) * NBLOCKS + n * SPLITS;
            for (int b = 0; b < SPLITS; ++b) s += p[b];
            q[k] = s;
        }
        const float sum1 = q[0], sum2 = q[1];
        const float e1 = q[2], e2 = q[3];
        const float norm = sqrtf(e1 * e2);
        // mean(full_xcorr)/norm + penalty
        loss += (sum1 * sum2) * inv_len / norm + 1.0f - sqrtf(e1 + e2) / norm;

        // sparsity: channels 1 and 2
        for (int c = 0; c < 2; ++c) {
            const float si = SPARSITY_A * q[4 + c];
            const float st = SPARSITY_A * q[6 + c];
            loss += fabsf(si - st) / (st + 1.0f);
        }
    }

    out[0] = loss;
}

// ---------------------------------------------------------------------------
extern "C" void kernel_launch(void* const* d_in, const int* in_sizes, int n_in,
                              void* d_out, int out_size, void* d_ws, size_t ws_size,
                              hipStream_t stream) {
    (void)in_sizes; (void)n_in; (void)out_size; (void)ws_size;
    const float* inp = (const float*)d_in[0];   // (8, 8192, 3) f32
    const float* tgt = (const float*)d_in[1];   // (8, 8192, 3) f32
    const float* cw  = (const float*)d_in[2];   // (3,) f32
    const float* pw  = (const float*)d_in[3];   // (3,) f32
    float* out = (float*)d_out;
    float* ws  = (float*)d_ws;                  // NQ * NBLOCKS floats used

    loss_partials_kernel<<<NBLOCKS, TPB, 0, stream>>>(inp, tgt, pw, ws);
    loss_finalize_kernel<<<1, 32, 0, stream>>>(ws, cw, out);
}